// EdgePoolEncoder_25039659336314
// MI455X (gfx1250) — compile-verified
//
#include <hip/hip_runtime.h>
#include <math.h>

#define N_NODES 32768
#define E_EDGES 262144
#define NP1     32769
#define NEGS    0.2f
#define EPSV    1e-5f
#define ADD_SCOREV 0.5f

typedef float v2f __attribute__((ext_vector_type(2)));
typedef float v8f __attribute__((ext_vector_type(8)));

// ---------- helpers ----------
__device__ __forceinline__ void atomicMaxF(float* addr, float v) {
  if (v >= 0.0f) atomicMax((int*)addr, __float_as_int(v));
  else           atomicMin((unsigned int*)addr, __float_as_uint(v));
}

__global__ void k_fill_u32(unsigned* p, unsigned v, int n) {
  int i = blockIdx.x * 256 + threadIdx.x;
  if (i < n) p[i] = v;
}

// ---------- pool score: a = x@pw[:256], b = x@pw[256:] ----------
__global__ void k_matvec_ab(const float* __restrict__ x, const float* __restrict__ pw,
                            float* __restrict__ a, float* __restrict__ b) {
  int t = threadIdx.x, lane = t & 31;
  int row = blockIdx.x * 8 + (t >> 5);
  const float* xr = x + row * 256;
  float sa = 0.f, sb = 0.f;
#pragma unroll
  for (int i = 0; i < 8; ++i) {
    int c = lane + i * 32;
    float v = xr[c];
    sa += v * pw[c];
    sb += v * pw[256 + c];
  }
  for (int o = 16; o > 0; o >>= 1) { sa += __shfl_xor(sa, o); sb += __shfl_xor(sb, o); }
  if (lane == 0) { a[row] = sa; b[row] = sb; }
}

__global__ void k_edge_raw(const long long* __restrict__ ei, const float* __restrict__ a,
                           const float* __restrict__ b, const float* __restrict__ pb,
                           float* __restrict__ raw, float* __restrict__ mx) {
  int e = blockIdx.x * 256 + threadIdx.x;
  int s = (int)ei[e], d = (int)ei[E_EDGES + e];
  float r = a[s] + b[d] + pb[0];
  raw[e] = r;
  atomicMaxF(&mx[d], r);
}

__global__ void k_edge_exp(const long long* __restrict__ ei, float* __restrict__ raw,
                           const float* __restrict__ mx, float* __restrict__ sm) {
  int e = blockIdx.x * 256 + threadIdx.x;
  int d = (int)ei[E_EDGES + e];
  float ex = expf(raw[e] - mx[d]);
  raw[e] = ex;
  atomicAdd(&sm[d], ex);
}

__global__ void k_edge_score_key(const long long* __restrict__ ei, const float* __restrict__ raw,
                                 const float* __restrict__ sm, float* __restrict__ score,
                                 unsigned long long* __restrict__ keys) {
  int e = blockIdx.x * 256 + threadIdx.x;
  int d = (int)ei[E_EDGES + e];
  float sc = raw[e] / sm[d] + ADD_SCOREV;
  score[e] = sc;
  unsigned u = __float_as_uint(sc);
  u = (u & 0x80000000u) ? ~u : (u | 0x80000000u);   // ascending u <=> ascending float
  keys[e] = ((unsigned long long)(~u) << 32) | (unsigned)e;  // asc key = score desc, idx asc
}

// ---------- bitonic sorts (E = 2^18 exactly) ----------
__global__ void k_bitonic_u64(unsigned long long* __restrict__ kk, unsigned j, unsigned k) {
  unsigned i = blockIdx.x * 256 + threadIdx.x;
  unsigned ixj = i ^ j;
  if (ixj > i) {
    unsigned long long A = kk[i], B = kk[ixj];
    bool up = ((i & k) == 0);
    if ((A > B) == up) { kk[i] = B; kk[ixj] = A; }
  }
}
__global__ void k_bitonic_u32(unsigned* __restrict__ kk, unsigned j, unsigned k) {
  unsigned i = blockIdx.x * 256 + threadIdx.x;
  unsigned ixj = i ^ j;
  if (ixj > i) {
    unsigned A = kk[i], B = kk[ixj];
    bool up = ((i & k) == 0);
    if ((A > B) == up) { kk[i] = B; kk[ixj] = A; }
  }
}

// ---------- greedy matching (sequential, LDS bitmap) ----------
__global__ void k_init_cluster(int* __restrict__ cluster, float* __restrict__ nscore,
                               int* __restrict__ cnt) {
  int i = blockIdx.x * 256 + threadIdx.x;
  if (i < N_NODES) { cluster[i] = i; nscore[i] = 1.0f; cnt[i] = 0; }
}

__global__ void k_match(const unsigned long long* __restrict__ keys, const long long* __restrict__ ei,
                        const float* __restrict__ score, int* __restrict__ cluster,
                        float* __restrict__ nscore) {
  __shared__ unsigned mb[N_NODES / 32];
  for (int i = threadIdx.x; i < N_NODES / 32; i += blockDim.x) mb[i] = 0u;
  __syncthreads();
  if (threadIdx.x == 0) {
    for (int i = 0; i < E_EDGES; ++i) {
      int e = (int)(keys[i] & 0xffffffffu);
      int s = (int)ei[e], d = (int)ei[E_EDGES + e];
      if (!((mb[s >> 5] >> (s & 31)) & 1u) && !((mb[d >> 5] >> (d & 31)) & 1u)) {
        mb[s >> 5] |= (1u << (s & 31));
        mb[d >> 5] |= (1u << (d & 31));
        int rep = s < d ? s : d;
        cluster[s] = rep; cluster[d] = rep;
        nscore[rep] = score[e];
      }
    }
  }
}

__global__ void k_counts(const int* __restrict__ cluster, int* __restrict__ cnt) {
  int i = blockIdx.x * 256 + threadIdx.x;
  if (i < N_NODES) atomicAdd(&cnt[cluster[i]], 1);
}

__global__ void k_xpool_scatter(const float* __restrict__ x, const int* __restrict__ cluster,
                                float* __restrict__ xp) {
  int g = blockIdx.x * 256 + threadIdx.x;   // over N*256
  int row = g >> 8, c = g & 255;
  atomicAdd(&xp[cluster[row] * 256 + c], x[g]);
}
__global__ void k_xpool_scale(float* __restrict__ xp, const float* __restrict__ nscore) {
  int g = blockIdx.x * 256 + threadIdx.x;   // over N*256
  xp[g] *= nscore[g >> 8];
}

// ---------- pooled-edge dedup ----------
__global__ void k_key2(const long long* __restrict__ ei, const int* __restrict__ cluster,
                       unsigned* __restrict__ k2) {
  int e = blockIdx.x * 256 + threadIdx.x;
  k2[e] = (unsigned)cluster[(int)ei[e]] * (unsigned)NP1 + (unsigned)cluster[(int)ei[E_EDGES + e]];
}
__global__ void k_dedup(const unsigned* __restrict__ k2, int* __restrict__ psrc,
                        int* __restrict__ pdst) {
  int e = blockIdx.x * 256 + threadIdx.x;
  unsigned k = k2[e];
  bool dup = (e > 0) && (k2[e - 1] == k);
  psrc[e] = dup ? N_NODES : (int)(k / NP1);
  pdst[e] = dup ? N_NODES : (int)(k % NP1);
}

// ---------- WMMA fp32 GEMM: Y[Mo x 256] = X[Mv x 256] @ W[256 x 256] (+bias) ----------
// Row index clamped to Mv-1 (garbage rows never stored since Mo <= Mv rows map 1:1).
// LDS holds a 64k x 64n chunk of W transposed [n][k], stride 68 -> conflict-free
// b64 B-fragment reads. W chunk is batch-loaded into registers before LDS stores,
// and the 16 A-fragments per chunk are prefetched into registers before the MMA loop.
__global__ __launch_bounds__(128) void k_gemm256(const float* __restrict__ X,
                                                 const float* __restrict__ W,
                                                 const float* __restrict__ bias,
                                                 float* __restrict__ Y, int Mv, int Mo) {
  __shared__ float wlds[64 * 68];
  const int tid  = threadIdx.x;
  const int wave = tid >> 5;
  const int lane = tid & 31;
  const int l15  = lane & 15;
  const int koff = (lane < 16) ? 0 : 2;
  const int m0   = blockIdx.x * 64 + wave * 16;
  const int n0g  = blockIdx.y * 64;

  v8f acc[4];
#pragma unroll
  for (int nt = 0; nt < 4; ++nt)
#pragma unroll
    for (int i = 0; i < 8; ++i) acc[nt][i] = 0.0f;

  const int rr = m0 + l15;
  const int r  = (rr < Mv) ? rr : (Mv - 1);   // clamp: no divergence in hot loop
  const float* xrow = X + r * 256 + koff;

  for (int kc = 0; kc < 256; kc += 64) {
    // ---- batch-load W chunk into registers, then stage into LDS ----
    float4 wreg[8];
#pragma unroll
    for (int i = 0; i < 8; ++i) {
      int idx = i * 128 + tid;          // 0..1023 float4s
      int kr = idx >> 4, nq = idx & 15;
      wreg[i] = *reinterpret_cast<const float4*>(W + (kc + kr) * 256 + n0g + nq * 4);
    }
#pragma unroll
    for (int i = 0; i < 8; ++i) {
      int idx = i * 128 + tid;
      int kr = idx >> 4, nb = (idx & 15) * 4;
      wlds[(nb + 0) * 68 + kr] = wreg[i].x;
      wlds[(nb + 1) * 68 + kr] = wreg[i].y;
      wlds[(nb + 2) * 68 + kr] = wreg[i].z;
      wlds[(nb + 3) * 68 + kr] = wreg[i].w;
    }

    // ---- prefetch this chunk's 16 A-fragments (unconditional) ----
    v2f areg[16];
#pragma unroll
    for (int kq = 0; kq < 16; ++kq) {
      float2 t = *reinterpret_cast<const float2*>(xrow + kc + kq * 4);
      areg[kq][0] = t.x; areg[kq][1] = t.y;
    }

    __syncthreads();
#pragma unroll
    for (int kq = 0; kq < 16; ++kq) {
      int k0 = kq * 4;
#pragma unroll
      for (int nt = 0; nt < 4; ++nt) {
        int nl = nt * 16 + l15;
        const float* wp = &wlds[nl * 68 + k0 + koff];
        v2f bv; bv[0] = wp[0]; bv[1] = wp[1];
        acc[nt] = __builtin_amdgcn_wmma_f32_16x16x4_f32(false, areg[kq], false, bv,
                                                        (short)0, acc[nt], false, false);
      }
    }
    __syncthreads();
  }

  const int rbase = m0 + ((lane < 16) ? 0 : 8);
#pragma unroll
  for (int nt = 0; nt < 4; ++nt) {
    int col = n0g + nt * 16 + l15;
    float bi = bias ? bias[col] : 0.0f;
#pragma unroll
    for (int i = 0; i < 8; ++i) {
      int row = rbase + i;
      if (row < Mo) Y[row * 256 + col] = acc[nt][i] + bi;
    }
  }
}

// ---------- GATv2 attention ----------
__global__ void k_att_logit(const float* __restrict__ xl, const float* __restrict__ xr,
                            const int* __restrict__ psrc, const int* __restrict__ pdst,
                            const float* __restrict__ att, float* __restrict__ logit,
                            float* __restrict__ emax) {
  int t = threadIdx.x, lane = t & 31;
  int e = blockIdx.x * 8 + (t >> 5);
  int ps = psrc[e], pd = pdst[e];
  const float* xa = xl + ps * 256;
  const float* xb = xr + pd * 256;
  int h = lane >> 3;
  float s = 0.f;
#pragma unroll
  for (int c = 0; c < 8; ++c) {
    int ch = lane * 8 + c;
    float v = xa[ch] + xb[ch];
    v = v > 0.f ? v : NEGS * v;
    s += v * att[h * 64 + (ch & 63)];
  }
  s += __shfl_xor(s, 4); s += __shfl_xor(s, 2); s += __shfl_xor(s, 1);
  if ((lane & 7) == 0) {
    logit[e * 4 + h] = s;
    atomicMaxF(&emax[pd * 4 + h], s);
  }
}

__global__ void k_att_expsum(const int* __restrict__ pdst, float* __restrict__ logit,
                             const float* __restrict__ emax, float* __restrict__ esum) {
  int g = blockIdx.x * 256 + threadIdx.x;   // over E*4
  int e = g >> 2, h = g & 3;
  int pd = pdst[e];
  float ex = expf(logit[g] - emax[pd * 4 + h]);
  logit[g] = ex;
  atomicAdd(&esum[pd * 4 + h], ex);
}

__global__ void k_att_out(const float* __restrict__ xl, const int* __restrict__ psrc,
                          const int* __restrict__ pdst, const float* __restrict__ logit,
                          const float* __restrict__ esum, float* __restrict__ out) {
  int t = threadIdx.x, lane = t & 31;
  int e = blockIdx.x * 8 + (t >> 5);
  int ps = psrc[e], pd = pdst[e];
  int h = lane >> 3;
  float alpha = logit[e * 4 + h] / esum[pd * 4 + h];
  const float* xa = xl + ps * 256;
  float* orow = out + pd * 256;
#pragma unroll
  for (int c = 0; c < 8; ++c) {
    int ch = lane * 8 + c;
    atomicAdd(&orow[ch], xa[ch] * alpha);
  }
}

// ---------- graph layer-norm + exact GELU ----------
__global__ void k_ln_init(double* s, int* vc) { s[0] = 0.0; s[1] = 0.0; *vc = 0; }

__global__ void k_ln_reduce(const float* __restrict__ h, const int* __restrict__ cnt,
                            double* __restrict__ s) {
  __shared__ double sh1[256], sh2[256];
  int t = threadIdx.x;
  int base = blockIdx.x * 2048;
  double l1 = 0.0, l2 = 0.0;
#pragma unroll
  for (int i = 0; i < 8; ++i) {
    int g = base + i * 256 + t;
    float v = h[g];
    if (cnt[g >> 8] > 0) { l1 += (double)v; l2 += (double)v * (double)v; }
  }
  sh1[t] = l1; sh2[t] = l2; __syncthreads();
  for (int o = 128; o > 0; o >>= 1) {
    if (t < o) { sh1[t] += sh1[t + o]; sh2[t] += sh2[t + o]; }
    __syncthreads();
  }
  if (t == 0) { atomicAdd(&s[0], sh1[0]); atomicAdd(&s[1], sh2[0]); }
}

__global__ void k_valid_count(const int* __restrict__ cnt, int* __restrict__ vc) {
  int i = blockIdx.x * 256 + threadIdx.x;
  if (i < N_NODES && cnt[i] > 0) atomicAdd(vc, 1);
}

__global__ void k_ln_fin(const double* __restrict__ s, const int* __restrict__ vc,
                         float* __restrict__ mi) {
  double c = (double)(*vc) * 256.0;
  double mean = s[0] / c;
  double var = s[1] / c - mean * mean;
  if (var < 0.0) var = 0.0;
  mi[0] = (float)mean;
  mi[1] = (float)(1.0 / (sqrt(var) + (double)EPSV));
}

__global__ void k_ln_gelu(float* __restrict__ h, const float* __restrict__ mi,
                          const float* __restrict__ w, const float* __restrict__ b) {
  int g = blockIdx.x * 256 + threadIdx.x;   // over N*256
  int c = g & 255;
  float v = (h[g] - mi[0]) * mi[1] * w[c] + b[c];
  h[g] = 0.5f * v * (1.0f + erff(v * 0.70710678118654752f));
}

// ---------- final gather: out = (xt / new_score)[cluster] ----------
__global__ void k_final(const float* __restrict__ xt, const int* __restrict__ cluster,
                        const float* __restrict__ nscore, float* __restrict__ out) {
  int g = blockIdx.x * 256 + threadIdx.x;   // over N*256
  int j = cluster[g >> 8];
  out[g] = xt[j * 256 + (g & 255)] / nscore[j];
}

// ---------------------------------------------------------------------------
extern "C" void kernel_launch(void* const* d_in, const int* in_sizes, int n_in,
                              void* d_out, int out_size, void* d_ws, size_t ws_size,
                              hipStream_t stream) {
  (void)in_sizes; (void)n_in; (void)out_size; (void)ws_size;
  const float*     x      = (const float*)d_in[0];
  const long long* ei     = (const long long*)d_in[1];
  const float*     pool_w = (const float*)d_in[4];
  const float*     pool_b = (const float*)d_in[5];
  const float*     w_l0   = (const float*)d_in[6];
  const float*     w_r0   = (const float*)d_in[7];
  const float*     att0   = (const float*)d_in[8];
  const float*     w_l1   = (const float*)d_in[9];
  const float*     w_r1   = (const float*)d_in[10];
  const float*     att1   = (const float*)d_in[11];
  const float*     ln_w0  = (const float*)d_in[12];
  const float*     ln_b0  = (const float*)d_in[13];
  const float*     ln_w1  = (const float*)d_in[14];
  const float*     ln_b1  = (const float*)d_in[15];
  const float*     lin_w  = (const float*)d_in[16];
  const float*     lin_b  = (const float*)d_in[17];
  float* out = (float*)d_out;

  // ---- workspace layout ----
  char* ws = (char*)d_ws;
  size_t off = 0;
  auto take = [&](size_t bytes) { size_t o = off; off += (bytes + 255) & ~(size_t)255; return o; };
  const size_t PB = (size_t)NP1 * 256 * 4;
  float*              xp      = (float*)(ws + take(PB));            // x_pool, later layer-1 output
  float*              xl      = (float*)(ws + take(PB));            // xl, later x_trans
  float*              xr      = (float*)(ws + take(PB));
  float*              hbuf    = (float*)(ws + take(PB));            // layer-0 output
  unsigned long long* skeys   = (unsigned long long*)(ws + take((size_t)E_EDGES * 8));
  float*              rawbuf  = (float*)(ws + take((size_t)E_EDGES * 4));   // raw/ex, reused as key2
  float*              score   = (float*)(ws + take((size_t)E_EDGES * 4));
  float*              logit   = (float*)(ws + take((size_t)E_EDGES * 16));
  int*                psrc    = (int*)(ws + take((size_t)E_EDGES * 4));
  int*                pdst    = (int*)(ws + take((size_t)E_EDGES * 4));
  float*              va      = (float*)(ws + take((size_t)N_NODES * 4));
  float*              vb      = (float*)(ws + take((size_t)N_NODES * 4));
  float*              mx      = (float*)(ws + take((size_t)N_NODES * 4));
  float*              sm      = (float*)(ws + take((size_t)N_NODES * 4));
  int*                cluster = (int*)(ws + take((size_t)N_NODES * 4));
  float*              nscore  = (float*)(ws + take((size_t)N_NODES * 4));
  int*                cnt     = (int*)(ws + take((size_t)N_NODES * 4));
  float*              emax    = (float*)(ws + take((size_t)NP1 * 4 * 4));
  float*              esum    = (float*)(ws + take((size_t)NP1 * 4 * 4));
  double*             sums    = (double*)(ws + take(64));
  int*                vc      = (int*)((char*)sums + 16);
  float*              mi      = (float*)((char*)sums + 32);

  const int EB = E_EDGES / 256;            // 1024
  const int NB = N_NODES / 256;            // 128
  const int GB = N_NODES;                  // N*256/256 = 32768 blocks of 256
  const int PW = (int)(PB / 4);            // (N+1)*256 floats

  // ---- edge pooling scores + softmax(dst) ----
  k_matvec_ab<<<N_NODES / 8, 256, 0, stream>>>(x, pool_w, va, vb);
  k_fill_u32<<<(N_NODES + 255) / 256, 256, 0, stream>>>((unsigned*)mx, 0xFF800000u, N_NODES);
  k_fill_u32<<<(N_NODES + 255) / 256, 256, 0, stream>>>((unsigned*)sm, 0u, N_NODES);
  k_edge_raw<<<EB, 256, 0, stream>>>(ei, va, vb, pool_b, rawbuf, mx);
  k_edge_exp<<<EB, 256, 0, stream>>>(ei, rawbuf, mx, sm);
  k_edge_score_key<<<EB, 256, 0, stream>>>(ei, rawbuf, sm, score, skeys);

  // ---- stable argsort by (score desc, idx asc) ----
  for (unsigned k = 2; k <= E_EDGES; k <<= 1)
    for (unsigned j = k >> 1; j > 0; j >>= 1)
      k_bitonic_u64<<<EB, 256, 0, stream>>>(skeys, j, k);

  // ---- greedy matching + clustering ----
  k_init_cluster<<<NB, 256, 0, stream>>>(cluster, nscore, cnt);
  k_match<<<1, 256, 0, stream>>>(skeys, ei, score, cluster, nscore);
  k_counts<<<NB, 256, 0, stream>>>(cluster, cnt);
  k_fill_u32<<<(PW + 255) / 256, 256, 0, stream>>>((unsigned*)xp, 0u, PW);
  k_xpool_scatter<<<GB, 256, 0, stream>>>(x, cluster, xp);
  k_xpool_scale<<<GB, 256, 0, stream>>>(xp, nscore);

  // ---- pooled edges, dedup via 32-bit key sort ----
  unsigned* key2 = (unsigned*)rawbuf;
  k_key2<<<EB, 256, 0, stream>>>(ei, cluster, key2);
  for (unsigned k = 2; k <= E_EDGES; k <<= 1)
    for (unsigned j = k >> 1; j > 0; j >>= 1)
      k_bitonic_u32<<<EB, 256, 0, stream>>>(key2, j, k);
  k_dedup<<<EB, 256, 0, stream>>>(key2, psrc, pdst);

  dim3 gemm_blk(128);
  dim3 gemm_grd((NP1 + 63) / 64, 4);       // 513 x 4
  dim3 gemm_grd_n((N_NODES + 63) / 64, 4); // 512 x 4

  // ---- GATv2 layer 0 (xp row N is zero, so xl/xr row N computed = 0) ----
  k_gemm256<<<gemm_grd, gemm_blk, 0, stream>>>(xp, w_l0, nullptr, xl, NP1, NP1);
  k_gemm256<<<gemm_grd, gemm_blk, 0, stream>>>(xp, w_r0, nullptr, xr, NP1, NP1);
  k_fill_u32<<<(NP1 * 4 + 255) / 256, 256, 0, stream>>>((unsigned*)emax, 0xFF800000u, NP1 * 4);
  k_fill_u32<<<(NP1 * 4 + 255) / 256, 256, 0, stream>>>((unsigned*)esum, 0u, NP1 * 4);
  k_fill_u32<<<(PW + 255) / 256, 256, 0, stream>>>((unsigned*)hbuf, 0u, PW);
  k_att_logit<<<E_EDGES / 8, 256, 0, stream>>>(xl, xr, psrc, pdst, att0, logit, emax);
  k_att_expsum<<<E_EDGES * 4 / 256, 256, 0, stream>>>(pdst, logit, emax, esum);
  k_att_out<<<E_EDGES / 8, 256, 0, stream>>>(xl, psrc, pdst, logit, esum, hbuf);
  k_ln_init<<<1, 1, 0, stream>>>(sums, vc);
  k_ln_reduce<<<N_NODES * 256 / 2048, 256, 0, stream>>>(hbuf, cnt, sums);
  k_valid_count<<<NB, 256, 0, stream>>>(cnt, vc);
  k_ln_fin<<<1, 1, 0, stream>>>(sums, vc, mi);
  k_ln_gelu<<<GB, 256, 0, stream>>>(hbuf, mi, ln_w0, ln_b0);

  // ---- GATv2 layer 1 (store rows < N, then explicitly zero row N of xl/xr) ----
  k_gemm256<<<gemm_grd, gemm_blk, 0, stream>>>(hbuf, w_l1, nullptr, xl, N_NODES, N_NODES);
  k_gemm256<<<gemm_grd, gemm_blk, 0, stream>>>(hbuf, w_r1, nullptr, xr, N_NODES, N_NODES);
  k_fill_u32<<<1, 256, 0, stream>>>((unsigned*)(xl + (size_t)N_NODES * 256), 0u, 256);
  k_fill_u32<<<1, 256, 0, stream>>>((unsigned*)(xr + (size_t)N_NODES * 256), 0u, 256);
  k_fill_u32<<<(NP1 * 4 + 255) / 256, 256, 0, stream>>>((unsigned*)emax, 0xFF800000u, NP1 * 4);
  k_fill_u32<<<(NP1 * 4 + 255) / 256, 256, 0, stream>>>((unsigned*)esum, 0u, NP1 * 4);
  k_fill_u32<<<(PW + 255) / 256, 256, 0, stream>>>((unsigned*)xp, 0u, PW);  // reuse xp as h2
  k_att_logit<<<E_EDGES / 8, 256, 0, stream>>>(xl, xr, psrc, pdst, att1, logit, emax);
  k_att_expsum<<<E_EDGES * 4 / 256, 256, 0, stream>>>(pdst, logit, emax, esum);
  k_att_out<<<E_EDGES / 8, 256, 0, stream>>>(xl, psrc, pdst, logit, esum, xp);
  k_ln_init<<<1, 1, 0, stream>>>(sums, vc);
  k_ln_reduce<<<N_NODES * 256 / 2048, 256, 0, stream>>>(xp, cnt, sums);
  k_valid_count<<<NB, 256, 0, stream>>>(cnt, vc);
  k_ln_fin<<<1, 1, 0, stream>>>(sums, vc, mi);
  k_ln_gelu<<<GB, 256, 0, stream>>>(xp, mi, ln_w1, ln_b1);

  // ---- final linear + unpool gather ----
  k_gemm256<<<gemm_grd_n, gemm_blk, 0, stream>>>(xp, lin_w, lin_b, xl, N_NODES, N_NODES);
  k_final<<<GB, 256, 0, stream>>>(xl, cluster, nscore, out);
}